// AFGNNlayer_62586263437745
// MI455X (gfx1250) — compile-verified
//
#include <hip/hip_runtime.h>

#define NN      100000
#define DD      128
#define NADJ    6
#define NNZT    600000
#define NHOPS   7
#define EPW     4   // edges per wave in spmm

typedef __attribute__((ext_vector_type(2))) float v2f;
typedef __attribute__((ext_vector_type(8))) float v8f;

// ---------------------------------------------------------------------------
// Kernel 1: softmax over the 7 mixing weights -> ws
// ---------------------------------------------------------------------------
__global__ void afgnn_mix_kernel(const float* __restrict__ lw,
                                 float* __restrict__ mix) {
    if (threadIdx.x == 0) {
        float m = lw[0];
        #pragma unroll
        for (int i = 1; i < NHOPS; ++i) m = fmaxf(m, lw[i]);
        float e[NHOPS];
        float s = 0.f;
        #pragma unroll
        for (int i = 0; i < NHOPS; ++i) { e[i] = __expf(lw[i] - m); s += e[i]; }
        float inv = 1.0f / s;
        #pragma unroll
        for (int i = 0; i < NHOPS; ++i) mix[i] = e[i] * inv;
    }
}

// ---------------------------------------------------------------------------
// Kernel 2: rep = mix[0] * input   (float4, fully coalesced, re-inits output)
// ---------------------------------------------------------------------------
__global__ void afgnn_rep_init_kernel(const float4* __restrict__ in4,
                                      float4* __restrict__ rep4,
                                      const float* __restrict__ mix) {
    const float m0 = mix[0];
    const int i = blockIdx.x * blockDim.x + threadIdx.x;   // 3.2M float4's
    float4 v = in4[i];
    v.x *= m0; v.y *= m0; v.z *= m0; v.w *= m0;
    rep4[i] = v;
}

// ---------------------------------------------------------------------------
// Kernel 3: edge-parallel SpMM scatter.  One wave per EPW (4) edges: each lane gathers
// a float4 of the source row (coalesced 512B) and issues 4 hardware f32
// atomic adds into the destination row.  Features live in L2 (51MB < 192MB).
// ---------------------------------------------------------------------------
__global__ void afgnn_spmm_kernel(const float* __restrict__ input,
                                  const long long* __restrict__ rows,
                                  const long long* __restrict__ cols,
                                  const float* __restrict__ vals,
                                  const float* __restrict__ mix,
                                  float* __restrict__ rep) {
    const int lane = threadIdx.x & 31;
    const long long wave =
        (long long)blockIdx.x * (blockDim.x >> 5) + (threadIdx.x >> 5);
    const long long e0 = wave * EPW;
    const long long etot = (long long)NADJ * NNZT;

    #pragma unroll
    for (int j = 0; j < EPW; ++j) {
        const long long e = e0 + j;
        if (e >= etot) return;
        const int a = (int)(e / NNZT);                 // which adjacency
        const long long r = rows[e];
        const long long c = cols[e];
        const float w = mix[a + 1] * vals[e];

        const float4 v = ((const float4*)(input + c * DD))[lane];
        float* dst = rep + r * DD + lane * 4;
        unsafeAtomicAdd(dst + 0, w * v.x);
        unsafeAtomicAdd(dst + 1, w * v.y);
        unsafeAtomicAdd(dst + 2, w * v.z);
        unsafeAtomicAdd(dst + 3, w * v.w);
    }
}

// ---------------------------------------------------------------------------
// Kernel 4: out = rep @ W + bias using V_WMMA_F32_16X16X4_F32 (full f32).
// W (64KB) staged in LDS; each wave owns one 16-row M tile, preloads the 32
// A fragments, then sweeps the 8 N tiles accumulating 32 WMMA steps each.
//
// ISA layouts (CDNA5 ISA):
//   A 16x4 f32 : lanes 0-15 rows M=0..15 hold K={0,1}; lanes 16-31 K={2,3}
//   B 4x16 f32 : mirrored (lane%16 = N, lane/16 selects K pair)
//   C/D 16x16  : VGPR v -> M=v (lanes 0-15) / M=v+8 (lanes 16-31), N=lane%16
// ---------------------------------------------------------------------------
__global__ __launch_bounds__(256)
void afgnn_gemm_kernel(const float* __restrict__ rep,
                       const float* __restrict__ W,
                       const float* __restrict__ bias,
                       float* __restrict__ out) {
    __shared__ float wlds[DD * DD];   // 64 KB of 320 KB

    {   // cooperative load of the 128x128 weight into LDS
        const float4* src = (const float4*)W;
        float4* dst = (float4*)wlds;
        for (int i = threadIdx.x; i < (DD * DD) / 4; i += blockDim.x)
            dst[i] = src[i];
    }
    __syncthreads();

    const int lane   = threadIdx.x & 31;
    const int waveId = threadIdx.x >> 5;
    const int mtile  = blockIdx.x * 8 + waveId;
    if (mtile >= NN / 16) return;

    const int l15 = lane & 15;
    const int kk  = (lane >> 4) << 1;     // 0 for lanes 0-15, 2 for 16-31
    const int m0  = mtile * 16;

    // Preload all 32 A fragments (16x4 f32 each: one float2 per lane)
    v2f a[32];
    const float* arow = rep + (size_t)(m0 + l15) * DD + kk;
    #pragma unroll
    for (int k = 0; k < 32; ++k)
        a[k] = *(const v2f*)(arow + 4 * k);

    #pragma unroll 1
    for (int nt = 0; nt < 8; ++nt) {
        const int n0 = nt * 16;
        v8f c = {};
        const float* bcol = wlds + n0 + l15;
        #pragma unroll
        for (int k = 0; k < 32; ++k) {
            v2f b;
            b.x = bcol[(4 * k + kk) * DD];
            b.y = bcol[(4 * k + kk + 1) * DD];
            c = __builtin_amdgcn_wmma_f32_16x16x4_f32(
                    false, a[k], false, b, (short)0, c, false, false);
        }
        const float bv = bias[n0 + l15];
        const int mrow = m0 + ((lane >> 4) << 3);
        #pragma unroll
        for (int v = 0; v < 8; ++v)
            out[(size_t)(mrow + v) * DD + n0 + l15] = c[v] + bv;
    }
}

// ---------------------------------------------------------------------------
extern "C" void kernel_launch(void* const* d_in, const int* in_sizes, int n_in,
                              void* d_out, int out_size, void* d_ws, size_t ws_size,
                              hipStream_t stream) {
    const float*     input    = (const float*)d_in[0];
    const long long* adj_rows = (const long long*)d_in[1];
    const long long* adj_cols = (const long long*)d_in[2];
    const float*     adj_vals = (const float*)d_in[3];
    const float*     weight   = (const float*)d_in[4];
    const float*     lw       = (const float*)d_in[5];
    const float*     bias     = (const float*)d_in[6];

    float* out = (float*)d_out;                    // [NN*DD]  (first tuple elem)
    float* rep = out + (size_t)NN * DD;            // [NN*DD]  (second tuple elem)
    float* mix = (float*)d_ws;                     // 7 floats of scratch

    // 1) softmax mixing weights
    afgnn_mix_kernel<<<1, 32, 0, stream>>>(lw, mix);

    // 2) rep = mix[0] * input   (12.8M f32 = 3.2M float4 = 12500 blocks)
    afgnn_rep_init_kernel<<<(NN * DD / 4) / 256, 256, 0, stream>>>(
        (const float4*)input, (float4*)rep, mix);

    // 3) atomic SpMM over 3.6M edges, one wave per EPW edges
    const long long waves = (long long)NADJ * NNZT / EPW;     // 900000
    afgnn_spmm_kernel<<<(int)(waves / 8), 256, 0, stream>>>(
        input, adj_rows, adj_cols, adj_vals, mix, rep);

    // 4) out = rep @ W + bias  (6250 M-tiles, 8 waves/block -> 782 blocks)
    afgnn_gemm_kernel<<<(NN / 16 + 7) / 8, 256, 0, stream>>>(
        rep, weight, bias, out);
}